// WassersteinLoss_53549652247366
// MI455X (gfx1250) — compile-verified
//
#include <hip/hip_runtime.h>
#include <math.h>
#include <utility>

// ---------------------------------------------------------------------------
// Debiased Sinkhorn divergence, B=4, N=M=4096, C=3, f32.
// Cost matrices recomputed on the fly from LDS-resident point clouds; the
// x.y^T term of each 16x16 tile comes from V_WMMA_F32_16X16X4_F32 (K=4,
// 4th component zero-padded). B operand is pre-staged in LDS in WMMA lane
// layout so the inner loop has no divergence (EXEC all-ones at every WMMA).
// ---------------------------------------------------------------------------

typedef __attribute__((ext_vector_type(2))) float v2f;
typedef __attribute__((ext_vector_type(8))) float v8f;

#define NPTS 4096
#define ROWS_PER_WG 128   // 8 waves * 16 rows

// softmin over rows of X against columns (points) of Y:
//   out[i] = -eps * log( sum_j exp( logw + pot[j]/eps - C_ij/eps ) )
// with C_ij = 0.5*(|x_i|^2 + |y_j|^2) - x_i.y_j.
// The row term 0.5|x_i|^2/eps cancels inside the logsumexp and is applied
// only at writeback. If oldv != nullptr: out[i] = 0.5*(oldv[i] + softmin).
__global__ __launch_bounds__(256)
void sinkhorn_softmin_kernel(const float* __restrict__ X,
                             const float* __restrict__ Y,
                             const float* __restrict__ pot,
                             const float* __restrict__ oldv,
                             float* __restrict__ out,
                             float eps, float log_w) {
    // B operand staged per half-wave: half 0 -> {y0,y1}, half 1 -> {y2,0}
    __shared__ float Bst[2 * NPTS * 2];  // 64 KB
    __shared__ float csm[NPTS];          // 16 KB: logw + pot/eps - 0.5|y|^2/eps

    const int b   = blockIdx.y;
    const int tid = threadIdx.x;
    const float inv_eps = 1.0f / eps;

    // ---- stage Y (WMMA-B layout) and the column vector into LDS ----
    const float* Yb   = Y + (size_t)b * NPTS * 3;
    const float* potb = pot ? (pot + b * NPTS) : nullptr;
    for (int j = tid; j < NPTS; j += 256) {
        float y0 = Yb[j * 3 + 0], y1 = Yb[j * 3 + 1], y2 = Yb[j * 3 + 2];
        Bst[j * 2 + 0] = y0;
        Bst[j * 2 + 1] = y1;
        Bst[NPTS * 2 + j * 2 + 0] = y2;
        Bst[NPTS * 2 + j * 2 + 1] = 0.0f;
        float c = log_w - 0.5f * (y0 * y0 + y1 * y1 + y2 * y2) * inv_eps;
        if (potb) c += potb[j] * inv_eps;
        csm[j] = c;
    }
    __syncthreads();

    // ---- per-wave 16-row strip ----
    const int wave    = tid >> 5;
    const int lane    = tid & 31;
    const int half    = lane >> 4;       // 0: lanes 0-15, 1: lanes 16-31
    const int l       = lane & 15;
    const int rowbase = blockIdx.x * ROWS_PER_WG + wave * 16;

    const float* Xb = X + (size_t)b * NPTS * 3;

    // A operand (16x4 f32): lanes 0-15 hold K=0,1 of row l; lanes 16-31 K=2,3.
    // Scaled by 1/eps so WMMA yields x.y/eps directly. K=3 is zero pad.
    const int am = rowbase + l;
    v2f A;
    A.x = Xb[am * 3 + half * 2] * inv_eps;          // half==0 -> x0, half==1 -> x2
    A.y = half ? 0.0f : (Xb[am * 3 + 1] * inv_eps); // half==0 -> x1, half==1 -> pad

    // per-lane row constants (only used at writeback)
    float r[8];
#pragma unroll
    for (int k = 0; k < 8; ++k) {
        int rm = rowbase + half * 8 + k;
        float x0 = Xb[rm * 3 + 0], x1 = Xb[rm * 3 + 1], x2 = Xb[rm * 3 + 2];
        r[k] = 0.5f * (x0 * x0 + x1 * x1 + x2 * x2) * inv_eps;
    }

    // uniform per-lane B base pointer (no divergence in the hot loops)
    const float* Bp = Bst + half * (NPTS * 2);

    // ---- pass 1: per-row running max of t_ij = (x.y)/eps + c_j ----
    float mk[8];
#pragma unroll
    for (int k = 0; k < 8; ++k) mk[k] = -3.4e38f;

    for (int j = 0; j < NPTS; j += 16) {
        const int jc = j + l;
        v2f Bv = *(const v2f*)(Bp + jc * 2);
        v8f acc = {};
        v8f d = __builtin_amdgcn_wmma_f32_16x16x4_f32(
            false, A, false, Bv, (short)0, acc, false, false);
        float cj = csm[jc];
#pragma unroll
        for (int k = 0; k < 8; ++k)
            mk[k] = fmaxf(mk[k], d[k] + cj);
    }
    // reduce max across the 16 lanes of each half (cols of the tile)
#pragma unroll
    for (int off = 1; off < 16; off <<= 1) {
#pragma unroll
        for (int k = 0; k < 8; ++k)
            mk[k] = fmaxf(mk[k], __shfl_xor(mk[k], off, 32));
    }

    // ---- pass 2: sum of exp(t - rowmax) ----
    float sk[8];
#pragma unroll
    for (int k = 0; k < 8; ++k) sk[k] = 0.0f;

    for (int j = 0; j < NPTS; j += 16) {
        const int jc = j + l;
        v2f Bv = *(const v2f*)(Bp + jc * 2);
        v8f acc = {};
        v8f d = __builtin_amdgcn_wmma_f32_16x16x4_f32(
            false, A, false, Bv, (short)0, acc, false, false);
        float cj = csm[jc];
#pragma unroll
        for (int k = 0; k < 8; ++k)
            sk[k] += __expf(d[k] + cj - mk[k]);
    }
#pragma unroll
    for (int off = 1; off < 16; off <<= 1) {
#pragma unroll
        for (int k = 0; k < 8; ++k)
            sk[k] += __shfl_xor(sk[k], off, 32);
    }

    // ---- writeback: lanes 0 and 16 own rows [rowbase+half*8, +8) ----
    if (l == 0) {
        const float* ob = oldv ? (oldv + b * NPTS) : nullptr;
        float* outb     = out + b * NPTS;
#pragma unroll
        for (int k = 0; k < 8; ++k) {
            int rm = rowbase + half * 8 + k;
            float v = -eps * (mk[k] - r[k] + __logf(sk[k]));
            if (ob) v = 0.5f * (ob[rm] + v);
            outb[rm] = v;
        }
    }
}

// loss = sum_b [ mean_i(F-Fx) + mean_j(G-Gy) ]  with N==M==4096
__global__ __launch_bounds__(256)
void sinkhorn_reduce_kernel(const float* __restrict__ F,
                            const float* __restrict__ G,
                            const float* __restrict__ Fx,
                            const float* __restrict__ Gy,
                            float* __restrict__ outp) {
    __shared__ float sm[256];
    float acc = 0.0f;
    for (int i = threadIdx.x; i < 4 * NPTS; i += 256)
        acc += F[i] + G[i] - Fx[i] - Gy[i];
    sm[threadIdx.x] = acc;
    __syncthreads();
    for (int s = 128; s > 0; s >>= 1) {
        if (threadIdx.x < s) sm[threadIdx.x] += sm[threadIdx.x + s];
        __syncthreads();
    }
    if (threadIdx.x == 0) outp[0] = sm[0] * (1.0f / (float)NPTS);
}

extern "C" void kernel_launch(void* const* d_in, const int* in_sizes, int n_in,
                              void* d_out, int out_size, void* d_ws, size_t ws_size,
                              hipStream_t stream) {
    (void)in_sizes; (void)n_in; (void)out_size; (void)ws_size;
    const float* x = (const float*)d_in[0];   // generated [4,64,64,3] -> [4,4096,3]
    const float* y = (const float*)d_in[1];   // target    [4,64,64,3] -> [4,4096,3]

    // eps annealing schedule (diameter 4, p=2, scaling 0.5, blur 1e-3,
    // plus 3 refinement steps at blur^2) — matches the reference.
    float epsv[32];
    int ne = 0;
    {
        double e = 4.0;
        while (e > 1e-3) { epsv[ne++] = (float)(e * e); e *= 0.5; }
        for (int i = 0; i < 3; ++i) epsv[ne++] = 1e-6f;
    }
    const float log_w = -logf((float)NPTS);   // uniform weights, N == M

    // workspace: 8 potential buffers of B*N floats (double-buffered f,g,px,py)
    float* ws = (float*)d_ws;
    const int PSZ = 4 * NPTS;                 // 16384 floats per buffer
    float* f   = ws + 0 * PSZ;
    float* g   = ws + 1 * PSZ;
    float* px  = ws + 2 * PSZ;
    float* py  = ws + 3 * PSZ;
    float* fn  = ws + 4 * PSZ;
    float* gn  = ws + 5 * PSZ;
    float* pxn = ws + 6 * PSZ;
    float* pyn = ws + 7 * PSZ;

    dim3 grid(NPTS / ROWS_PER_WG, 4);
    dim3 block(256);

    auto softmin = [&](const float* Xp, const float* Yp, const float* potp,
                       const float* oldp, float* outp, float eps) {
        sinkhorn_softmin_kernel<<<grid, block, 0, stream>>>(
            Xp, Yp, potp, oldp, outp, eps, log_w);
    };

    // init at eps[0]
    softmin(x, y, nullptr, nullptr, f,  epsv[0]);
    softmin(y, x, nullptr, nullptr, g,  epsv[0]);
    softmin(x, x, nullptr, nullptr, px, epsv[0]);
    softmin(y, y, nullptr, nullptr, py, epsv[0]);

    // symmetric annealed updates
    for (int t = 1; t < ne; ++t) {
        float e = epsv[t];
        softmin(x, y, g,  f,  fn,  e);
        softmin(y, x, f,  g,  gn,  e);
        softmin(x, x, px, px, pxn, e);
        softmin(y, y, py, py, pyn, e);
        std::swap(f, fn); std::swap(g, gn);
        std::swap(px, pxn); std::swap(py, pyn);
    }

    // final extrapolation at blur^2
    const float ef = 1e-6f;
    softmin(x, y, g,  nullptr, fn,  ef);   // F
    softmin(y, x, f,  nullptr, gn,  ef);   // G
    softmin(x, x, px, nullptr, pxn, ef);   // Fx
    softmin(y, y, py, nullptr, pyn, ef);   // Gy

    sinkhorn_reduce_kernel<<<1, 256, 0, stream>>>(fn, gn, pxn, pyn, (float*)d_out);
}